// Decoder_62758062129336
// MI455X (gfx1250) — compile-verified
//
#include <hip/hip_runtime.h>
#include <math.h>

// Shapes from the reference
#define B_ 16
#define E_ 512
#define H_ 512
#define V_ 32000
#define L_ 512
#define N_ 20

typedef __attribute__((ext_vector_type(2))) float v2f;
typedef __attribute__((ext_vector_type(8))) float v8f;
typedef __attribute__((ext_vector_type(4))) unsigned int u32x4;
typedef __attribute__((ext_vector_type(8))) int i32x8;
typedef __attribute__((ext_vector_type(4))) int i32x4;

// D = A(16x4) * B(4x16) + C(16x16), fp32 WMMA (CDNA5)
__device__ __forceinline__ v8f wmma4(v2f a, v2f b, v8f c) {
  return __builtin_amdgcn_wmma_f32_16x16x4_f32(
      false, a, false, b, (short)0, c, false, false);
}

// Lane mapping (wave32) per ISA 7.12.2 for 32-bit operands:
//  A 16x4: lane = half*16 + m ; vgpr0 = A[m][k0+2*half], vgpr1 = A[m][k0+2*half+1]
//  B 4x16: lane = half*16 + n ; vgpr0 = B[2*half][n],    vgpr1 = B[2*half+1][n]
//  C/D 16x16: vgpr r -> row (r + 8*half), col = lane&15

// ---------------------------------------------------------------------------
// K1: h0[16,512] = concat(qn_t, g_t)[16,2048] @ fc_w[512,2048]^T + fc_b
// ---------------------------------------------------------------------------
__global__ __launch_bounds__(32) void k1_h0(
    const float* __restrict__ qn, const float* __restrict__ g,
    const float* __restrict__ fc_w, const float* __restrict__ fc_b,
    float* __restrict__ h0) {
  const int lane = threadIdx.x;
  const int half = lane >> 4, lm = lane & 15;
  const int j = blockIdx.x * 16 + lm;
  const float* wrow = fc_w + (size_t)j * (4 * H_);
  const int xrow = lm * (N_ * 2 * H_);  // q_n / g row stride (n=0 slice)
  v8f acc = {};
  for (int k0 = 0; k0 < 2048; k0 += 4) {
    const int k = k0 + 2 * half;
    v2f a, b;
    if (k < 1024) { a.x = qn[xrow + k];        a.y = qn[xrow + k + 1]; }
    else          { a.x = g [xrow + k - 1024]; a.y = g [xrow + k - 1023]; }
    b.x = wrow[k]; b.y = wrow[k + 1];
    acc = wmma4(a, b, acc);
  }
  const float bias = fc_b[j];
  for (int r = 0; r < 8; ++r)
    h0[(r + 8 * half) * H_ + j] = acc[r] + bias;
}

// ---------------------------------------------------------------------------
// K2: GRU cell. gi = embed[y] @ w_ih^T, gh = h0 @ w_hh^T (3 gate tiles each),
//     then elementwise r/z/n -> h[16,512]
// ---------------------------------------------------------------------------
__global__ __launch_bounds__(32) void k2_gru(
    const int* __restrict__ y, const float* __restrict__ embed,
    const float* __restrict__ w_ih, const float* __restrict__ w_hh,
    const float* __restrict__ b_ih, const float* __restrict__ b_hh,
    const float* __restrict__ h0, float* __restrict__ h) {
  const int lane = threadIdx.x;
  const int half = lane >> 4, lm = lane & 15;
  const int j = blockIdx.x * 16 + lm;
  const size_t xrow = (size_t)y[lm] * E_;

  const float* wi0 = w_ih + (size_t)(0 * H_ + j) * E_;
  const float* wi1 = w_ih + (size_t)(1 * H_ + j) * E_;
  const float* wi2 = w_ih + (size_t)(2 * H_ + j) * E_;
  const float* wh0 = w_hh + (size_t)(0 * H_ + j) * H_;
  const float* wh1 = w_hh + (size_t)(1 * H_ + j) * H_;
  const float* wh2 = w_hh + (size_t)(2 * H_ + j) * H_;

  v8f gi0 = {}, gi1 = {}, gi2 = {}, gh0 = {}, gh1 = {}, gh2 = {};
  for (int k0 = 0; k0 < 512; k0 += 4) {
    const int k = k0 + 2 * half;
    v2f ax, ah, b;
    ax.x = embed[xrow + k]; ax.y = embed[xrow + k + 1];
    ah.x = h0[lm * H_ + k]; ah.y = h0[lm * H_ + k + 1];
    b.x = wi0[k]; b.y = wi0[k + 1]; gi0 = wmma4(ax, b, gi0);
    b.x = wi1[k]; b.y = wi1[k + 1]; gi1 = wmma4(ax, b, gi1);
    b.x = wi2[k]; b.y = wi2[k + 1]; gi2 = wmma4(ax, b, gi2);
    b.x = wh0[k]; b.y = wh0[k + 1]; gh0 = wmma4(ah, b, gh0);
    b.x = wh1[k]; b.y = wh1[k + 1]; gh1 = wmma4(ah, b, gh1);
    b.x = wh2[k]; b.y = wh2[k + 1]; gh2 = wmma4(ah, b, gh2);
  }
  const float bi_r = b_ih[j],          bh_r = b_hh[j];
  const float bi_z = b_ih[H_ + j],     bh_z = b_hh[H_ + j];
  const float bi_n = b_ih[2 * H_ + j], bh_n = b_hh[2 * H_ + j];
  for (int r = 0; r < 8; ++r) {
    const int m = r + 8 * half;
    const float gr = gi0[r] + bi_r + gh0[r] + bh_r;
    const float gz = gi1[r] + bi_z + gh1[r] + bh_z;
    const float rr = 1.0f / (1.0f + expf(-gr));
    const float zz = 1.0f / (1.0f + expf(-gz));
    const float nn = tanhf(gi2[r] + bi_n + rr * (gh2[r] + bh_n));
    h[m * H_ + j] = (1.0f - zz) * nn + zz * h0[m * H_ + j];
  }
}

// ---------------------------------------------------------------------------
// K3a: u[16,1024] = h[16,512] @ w_la[512,1024]  (w_la stored (H, 2H))
// ---------------------------------------------------------------------------
__global__ __launch_bounds__(32) void k3a_u(
    const float* __restrict__ h, const float* __restrict__ w_la,
    float* __restrict__ u) {
  const int lane = threadIdx.x;
  const int half = lane >> 4, lm = lane & 15;
  const int d = blockIdx.x * 16 + lm;
  v8f acc = {};
  for (int k0 = 0; k0 < 512; k0 += 4) {
    const int k = k0 + 2 * half;
    v2f a, b;
    a.x = h[lm * H_ + k];                 a.y = h[lm * H_ + k + 1];
    b.x = w_la[(size_t)k * (2 * H_) + d]; b.y = w_la[(size_t)(k + 1) * (2 * H_) + d];
    acc = wmma4(a, b, acc);
  }
  for (int r = 0; r < 8; ++r)
    u[(r + 8 * half) * (2 * H_) + d] = acc[r];
}

// ---------------------------------------------------------------------------
// K3b: per-batch attention. scores[l] = q_t[b,l,:].u[b,:]; beta = softmax;
//      c = beta @ q_t; p_gen = leaky_relu(h.wh + c.wc + embed[y].wy, 0.2);
//      P_w[l] = p_gen + beta[l]*(1-p_gen)   (P_vocab == 1 exactly)
//
// u[b] (4KB) is staged into LDS by the Tensor Data Mover (TDM) from wave 0;
// scores use wave-per-row coalesced dots with a wave32 shuffle butterfly.
// ---------------------------------------------------------------------------
__global__ __launch_bounds__(256) void k3b_attn(
    const float* __restrict__ q, const float* __restrict__ u,
    const float* __restrict__ h, const int* __restrict__ y,
    const float* __restrict__ embed,
    const float* __restrict__ wh_w, const float* __restrict__ wc_w,
    const float* __restrict__ wy_w, float* __restrict__ Pw) {
  __shared__ float s_u[2 * H_];
  __shared__ float s_sc[L_];
  __shared__ float s_c[2 * H_];
  __shared__ float s_red[256];
  const int b = blockIdx.x, t = threadIdx.x;
  const int lane = t & 31, wave = t >> 5;  // 8 waves of 32
  const float* qb = q + (size_t)b * N_ * L_ * (2 * H_);  // q[b,0,:,:]

  // --- TDM: DMA u[b] (1024 f32) into s_u; issued once by wave 0 ---
  if (wave == 0) {
    const unsigned lds_addr = (unsigned)(uintptr_t)(&s_u[0]);
    const unsigned long long ga = (unsigned long long)(uintptr_t)(u + (size_t)b * (2 * H_));
    // D# group0: count=1 | lds_addr | global_addr[56:0] | type=2
    u32x4 g0 = { 1u, lds_addr, (unsigned)(ga & 0xFFFFFFFFu),
                 (unsigned)(((ga >> 32) & 0x01FFFFFFu) | (2u << 30)) };
    // D# group1: data_size=4B; tensor_dim0=1024; tensor_dim1=1; tile_dim0=1024;
    //            tile_dim1=1; tensor_dim0_stride=1024
    i32x8 g1 = { (int)(2u << 16),          // data_size=2 (4 bytes)
                 (int)(1024u << 16),       // tensor_dim0[15:0] << 16
                 (int)(1u << 16),          // tensor_dim0 hi=0 | tensor_dim1[15:0]<<16
                 (int)(1024u << 16),       // tensor_dim1 hi=0 | tile_dim0<<16
                 (int)1,                   // tile_dim1=1, tile_dim2=0
                 (int)1024,                // tensor_dim0_stride[31:0]
                 0, 0 };
    i32x4 gz = { 0, 0, 0, 0 };
#if __clang_major__ >= 23
    i32x8 gz8 = { 0, 0, 0, 0, 0, 0, 0, 0 };
    __builtin_amdgcn_tensor_load_to_lds(g0, g1, gz, gz, gz8, 0);
#else
    __builtin_amdgcn_tensor_load_to_lds(g0, g1, gz, gz, 0);
#endif
    __builtin_amdgcn_s_wait_tensorcnt(0);
  }
  __syncthreads();

  // scores: one wave per row (coalesced 128B reads), wave32 butterfly reduce
  for (int l = wave; l < L_; l += 8) {
    const float* qr = qb + (size_t)l * (2 * H_);
    float p = 0.0f;
    for (int d = lane; d < 2 * H_; d += 32) p += qr[d] * s_u[d];
    for (int off = 16; off > 0; off >>= 1) p += __shfl_xor(p, off, 32);
    if (lane == 0) s_sc[l] = p;
  }
  __syncthreads();

  // softmax over L_
  float mx = -INFINITY;
  for (int l = t; l < L_; l += 256) mx = fmaxf(mx, s_sc[l]);
  s_red[t] = mx; __syncthreads();
  for (int off = 128; off > 0; off >>= 1) {
    if (t < off) s_red[t] = fmaxf(s_red[t], s_red[t + off]);
    __syncthreads();
  }
  mx = s_red[0]; __syncthreads();
  float ps = 0.0f;
  for (int l = t; l < L_; l += 256) { float e = expf(s_sc[l] - mx); s_sc[l] = e; ps += e; }
  s_red[t] = ps; __syncthreads();
  for (int off = 128; off > 0; off >>= 1) {
    if (t < off) s_red[t] += s_red[t + off];
    __syncthreads();
  }
  const float inv = 1.0f / s_red[0]; __syncthreads();
  for (int l = t; l < L_; l += 256) s_sc[l] *= inv;  // beta
  __syncthreads();

  // context c[d] = sum_l beta[l] * q_t[b,l,d]  (d contiguous across threads)
  for (int d = t; d < 2 * H_; d += 256) {
    float cs = 0.0f;
    for (int l = 0; l < L_; ++l) cs += s_sc[l] * qb[(size_t)l * (2 * H_) + d];
    s_c[d] = cs;
  }
  __syncthreads();

  // p_gen
  float pg = 0.0f;
  const size_t xr = (size_t)y[b] * E_;
  for (int d = t; d < H_; d += 256) pg += h[b * H_ + d] * wh_w[d] + embed[xr + d] * wy_w[d];
  for (int d = t; d < 2 * H_; d += 256) pg += s_c[d] * wc_w[d];
  s_red[t] = pg; __syncthreads();
  for (int off = 128; off > 0; off >>= 1) {
    if (t < off) s_red[t] += s_red[t + off];
    __syncthreads();
  }
  float p = s_red[0];
  p = (p > 0.0f) ? p : 0.2f * p;  // leaky_relu(0.2)

  for (int l = t; l < L_; l += 256) Pw[b * L_ + l] = p + s_sc[l] * (1.0f - p);
}

// ---------------------------------------------------------------------------
// K4: pre[16,32000] = P_w[16,512] @ fco_w[32000,512]^T + fco_b
//     4 N-tiles per wave; fco_w (64MB) is streamed once -> non-temporal loads.
// ---------------------------------------------------------------------------
__global__ __launch_bounds__(32) void k4_out(
    const float* __restrict__ Pw, const float* __restrict__ fco_w,
    const float* __restrict__ fco_b, float* __restrict__ out) {
  const int lane = threadIdx.x;
  const int half = lane >> 4, lm = lane & 15;
  const int v0 = blockIdx.x * 64;
  const float* w0 = fco_w + (size_t)(v0 + 0 + lm) * L_;
  const float* w1 = fco_w + (size_t)(v0 + 16 + lm) * L_;
  const float* w2 = fco_w + (size_t)(v0 + 32 + lm) * L_;
  const float* w3 = fco_w + (size_t)(v0 + 48 + lm) * L_;
  v8f acc0 = {}, acc1 = {}, acc2 = {}, acc3 = {};
  for (int k0 = 0; k0 < L_; k0 += 4) {
    const int k = k0 + 2 * half;
    v2f a, b;
    a.x = Pw[lm * L_ + k]; a.y = Pw[lm * L_ + k + 1];
    b.x = __builtin_nontemporal_load(&w0[k]);
    b.y = __builtin_nontemporal_load(&w0[k + 1]);
    acc0 = wmma4(a, b, acc0);
    b.x = __builtin_nontemporal_load(&w1[k]);
    b.y = __builtin_nontemporal_load(&w1[k + 1]);
    acc1 = wmma4(a, b, acc1);
    b.x = __builtin_nontemporal_load(&w2[k]);
    b.y = __builtin_nontemporal_load(&w2[k + 1]);
    acc2 = wmma4(a, b, acc2);
    b.x = __builtin_nontemporal_load(&w3[k]);
    b.y = __builtin_nontemporal_load(&w3[k + 1]);
    acc3 = wmma4(a, b, acc3);
  }
  const float b0 = fco_b[v0 + 0 + lm], b1 = fco_b[v0 + 16 + lm];
  const float b2 = fco_b[v0 + 32 + lm], b3 = fco_b[v0 + 48 + lm];
  for (int r = 0; r < 8; ++r) {
    const int m = r + 8 * half;
    out[(size_t)m * V_ + v0 + 0 + lm] = acc0[r] + b0;
    out[(size_t)m * V_ + v0 + 16 + lm] = acc1[r] + b1;
    out[(size_t)m * V_ + v0 + 32 + lm] = acc2[r] + b2;
    out[(size_t)m * V_ + v0 + 48 + lm] = acc3[r] + b3;
  }
}

// ---------------------------------------------------------------------------
extern "C" void kernel_launch(void* const* d_in, const int* in_sizes, int n_in,
                              void* d_out, int out_size, void* d_ws, size_t ws_size,
                              hipStream_t stream) {
  const int*   y     = (const int*)  d_in[0];
  const float* q     = (const float*)d_in[1];
  const float* q_n   = (const float*)d_in[2];
  const float* g     = (const float*)d_in[3];
  const float* embed = (const float*)d_in[4];
  const float* fc_w  = (const float*)d_in[5];
  const float* fc_b  = (const float*)d_in[6];
  const float* w_ih  = (const float*)d_in[7];
  const float* w_hh  = (const float*)d_in[8];
  const float* b_ih  = (const float*)d_in[9];
  const float* b_hh  = (const float*)d_in[10];
  // d_in[11] w_ga, d_in[13] wv_w, d_in[14] wv_b: dead code (P_vocab == 1)
  const float* w_la  = (const float*)d_in[12];
  const float* wh_w  = (const float*)d_in[15];
  const float* wc_w  = (const float*)d_in[16];
  const float* wy_w  = (const float*)d_in[17];
  const float* fco_w = (const float*)d_in[18];
  const float* fco_b = (const float*)d_in[19];

  float* ws   = (float*)d_ws;
  float* ws_h0 = ws;            // 16*512
  float* ws_h  = ws + 8192;     // 16*512
  float* ws_u  = ws + 16384;    // 16*1024
  float* ws_Pw = ws + 32768;    // 16*512

  k1_h0  <<<H_ / 16,       32,  0, stream>>>(q_n, g, fc_w, fc_b, ws_h0);
  k2_gru <<<H_ / 16,       32,  0, stream>>>(y, embed, w_ih, w_hh, b_ih, b_hh, ws_h0, ws_h);
  k3a_u  <<<(2 * H_) / 16, 32,  0, stream>>>(ws_h, w_la, ws_u);
  k3b_attn<<<B_,           256, 0, stream>>>(q, ws_u, ws_h, y, embed, wh_w, wc_w, wy_w, ws_Pw);
  k4_out <<<V_ / 64,       32,  0, stream>>>(ws_Pw, fco_w, fco_b, (float*)d_out);
}